// GIN_DGL_Custom_55594056680298
// MI455X (gfx1250) — compile-verified
//
#include <hip/hip_runtime.h>
#include <hip/hip_bf16.h>

typedef __attribute__((ext_vector_type(2))) float v2f;
typedef __attribute__((ext_vector_type(8))) float v8f;
typedef int gv4i __attribute__((vector_size(16)));           // matches builtin param type
typedef __attribute__((address_space(1))) gv4i* gv4i_gptr;   // global (AS1)
typedef __attribute__((address_space(3))) gv4i* gv4i_lptr;   // LDS (AS3)

#define D 128
#define AS_STR 132      // padded LDS stride (floats) for A tile -> conflict-free frag reads
#define WP_STR 144      // padded LDS stride (float2) per K-pair row of packed W
#define BN_EPS 1e-5f

#define GIN_HAS_ASYNC __has_builtin(__builtin_amdgcn_global_load_async_to_lds_b128)

// ---------------- elementwise / graph kernels ----------------

__global__ __launch_bounds__(256) void copy4_kernel(const float4* __restrict__ x,
                                                    float4* __restrict__ y, long n4) {
  long i = (long)blockIdx.x * blockDim.x + threadIdx.x;
  long stride = (long)gridDim.x * blockDim.x;
  for (; i < n4; i += stride) y[i] = x[i];
}

__global__ __launch_bounds__(256) void zero_stats_kernel(float* __restrict__ s) {
  s[threadIdx.x] = 0.f;
}

// one wave32 per edge: lane L moves cols [4L, 4L+3] (512B per edge, coalesced)
__global__ __launch_bounds__(256) void edge_scatter_kernel(
    const float* __restrict__ h, const int* __restrict__ src,
    const int* __restrict__ dst, float* __restrict__ z, int ne) {
  int e = blockIdx.x * 8 + (threadIdx.x >> 5);
  if (e >= ne) return;
  int lane = threadIdx.x & 31;
  int s = src[e], d = dst[e];
  float4 v = ((const float4*)(h + (size_t)s * D))[lane];
  float* zp = z + (size_t)d * D + lane * 4;
  unsafeAtomicAdd(zp + 0, v.x);
  unsafeAtomicAdd(zp + 1, v.y);
  unsafeAtomicAdd(zp + 2, v.z);
  unsafeAtomicAdd(zp + 3, v.w);
}

// per-column sum & sum-of-squares over all rows (coalesced: 128 lanes = one row)
__global__ __launch_bounds__(256) void colstats_kernel(const float* __restrict__ x,
                                                       float* __restrict__ stats,
                                                       int nrows) {
  __shared__ float red[512];
  int col = threadIdx.x & 127;
  int half = threadIdx.x >> 7;
  float s = 0.f, s2 = 0.f;
  for (int r = blockIdx.x * 2 + half; r < nrows; r += gridDim.x * 2) {
    float v = x[(size_t)r * D + col];
    s += v;
    s2 += v * v;
  }
  red[threadIdx.x] = s;
  red[256 + threadIdx.x] = s2;
  __syncthreads();
  if (half == 0) {
    s += red[threadIdx.x + 128];
    s2 += red[256 + threadIdx.x + 128];
    unsafeAtomicAdd(&stats[col], s);
    unsafeAtomicAdd(&stats[128 + col], s2);
  }
}

// stats -> per-column affine: y = x*scale + shift  (folded BN)
__global__ __launch_bounds__(128) void finalize_bn_kernel(const float* __restrict__ stats,
                                                          const float* __restrict__ gamma,
                                                          const float* __restrict__ beta,
                                                          float* __restrict__ ss,
                                                          float inv_n) {
  int c = threadIdx.x;
  float mean = stats[c] * inv_n;
  float var = stats[128 + c] * inv_n - mean * mean;   // biased variance (matches ref)
  float sc = rsqrtf(var + BN_EPS) * gamma[c];
  ss[c] = sc;
  ss[128 + c] = beta[c] - mean * sc;
}

__global__ __launch_bounds__(256) void bnrelu_kernel(const float* __restrict__ x,
                                                     const float* __restrict__ ss,
                                                     float* __restrict__ y, long n4) {
  long i = (long)blockIdx.x * blockDim.x + threadIdx.x;
  long stride = (long)gridDim.x * blockDim.x;
  for (; i < n4; i += stride) {
    float4 v = ((const float4*)x)[i];
    int c = ((int)(i & 31)) * 4;
    v.x = fmaxf(fmaf(v.x, ss[c + 0], ss[128 + c + 0]), 0.f);
    v.y = fmaxf(fmaf(v.y, ss[c + 1], ss[128 + c + 1]), 0.f);
    v.z = fmaxf(fmaf(v.z, ss[c + 2], ss[128 + c + 2]), 0.f);
    v.w = fmaxf(fmaf(v.w, ss[c + 3], ss[128 + c + 3]), 0.f);
    ((float4*)y)[i] = v;
  }
}

// ---------------- WMMA fp32 GEMM: C[128-tile x 128] = op(A) @ W + bias ----------------
// Block = 256 threads (8 waves). Wave w computes rows [w*16, w*16+15] x all 128 cols,
// holding 8 accumulators (one per 16-col tile) so every register index is constant.
// A-frag layout per ISA (32-bit A 16x4): lanes 0-15 hold K={0,1}, lanes 16-31 K={2,3}.
// B-frag: lane holds {W[k0][col], W[k0+1][col]} -> W staged K-pair-interleaved in LDS
// so each B fragment is one contiguous, aligned 8-byte ds_load (no re-pairing movs).
// C/D layout: VGPR r -> M=r (lanes 0-15), M=8+r (lanes 16-31); N = lane&15.
template <bool PRO_BN, bool EPI_RELU>
__global__ __launch_bounds__(256) void gemm128_wmma(
    const float* __restrict__ A, const float* __restrict__ W,
    const float* __restrict__ bias, const float* __restrict__ ss,
    float* __restrict__ C, int nrows) {
  extern __shared__ float lds[];
  float* Ws = lds;                        // packed: [64 k-pairs][WP_STR float2]
  float* As = lds + 64 * WP_STR * 2;      // [128][AS_STR]
  const int tid = threadIdx.x;
  const int wave = tid >> 5;
  const int lane = tid & 31;
  const int rowBase = blockIdx.x * 128;

  // stage W into LDS, K-pair interleaved: Ws[2*(k2*WP_STR + c) + (k&1)] = W[k][c]
  for (int i = tid; i < 64 * 32; i += 256) {
    int k2 = i >> 5;
    int c4 = i & 31;
    float4 w0 = ((const float4*)W)[(2 * k2) * 32 + c4];
    float4 w1 = ((const float4*)W)[(2 * k2 + 1) * 32 + c4];
    float* dp = Ws + 2 * (k2 * WP_STR + c4 * 4);
    dp[0] = w0.x; dp[1] = w1.x; dp[2] = w0.y; dp[3] = w1.y;
    dp[4] = w0.z; dp[5] = w1.z; dp[6] = w0.w; dp[7] = w1.w;
  }

  // stage A tile into LDS (async DMA path when no prologue transform is needed)
  if (PRO_BN) {
    for (int i = tid; i < D * (D / 4); i += 256) {
      int r = i >> 5;
      int c4 = i & 31;
      int gr = rowBase + r;
      float4 v = make_float4(0.f, 0.f, 0.f, 0.f);
      if (gr < nrows) v = ((const float4*)(A + (size_t)gr * D))[c4];
      int c = c4 * 4;
      v.x = fmaxf(fmaf(v.x, ss[c + 0], ss[128 + c + 0]), 0.f);
      v.y = fmaxf(fmaf(v.y, ss[c + 1], ss[128 + c + 1]), 0.f);
      v.z = fmaxf(fmaf(v.z, ss[c + 2], ss[128 + c + 2]), 0.f);
      v.w = fmaxf(fmaf(v.w, ss[c + 3], ss[128 + c + 3]), 0.f);
      float* dp = As + r * AS_STR + c4 * 4;
      dp[0] = v.x; dp[1] = v.y; dp[2] = v.z; dp[3] = v.w;
    }
  } else {
#if GIN_HAS_ASYNC
    for (int i = tid; i < D * (D / 4); i += 256) {
      int r = i >> 5;
      int c4 = i & 31;
      int gr = rowBase + r;
      float* lp = As + r * AS_STR + c4 * 4;
      if (gr < nrows) {
        const float* gp = A + (size_t)gr * D + c4 * 4;
        __builtin_amdgcn_global_load_async_to_lds_b128(
            (gv4i_gptr)(gv4i*)gp, (gv4i_lptr)(gv4i*)lp, 0, 0);
      } else {
        lp[0] = 0.f; lp[1] = 0.f; lp[2] = 0.f; lp[3] = 0.f;
      }
    }
#if __has_builtin(__builtin_amdgcn_s_wait_asynccnt)
    __builtin_amdgcn_s_wait_asynccnt(0);
#else
    asm volatile("s_wait_asynccnt 0" ::: "memory");
#endif
#else
    for (int i = tid; i < D * (D / 4); i += 256) {
      int r = i >> 5;
      int c4 = i & 31;
      int gr = rowBase + r;
      float4 v = make_float4(0.f, 0.f, 0.f, 0.f);
      if (gr < nrows) v = ((const float4*)(A + (size_t)gr * D))[c4];
      float* dp = As + r * AS_STR + c4 * 4;
      dp[0] = v.x; dp[1] = v.y; dp[2] = v.z; dp[3] = v.w;
    }
#endif
  }
  __syncthreads();

  const int m = lane & 15;
  const int kh = lane >> 4;            // 0: K pair {0,1} of each 4-step; 1: pair {2,3}

  // 8 accumulators (64 VGPRs), all indices compile-time constant -> no spills
  v8f acc[8];
#pragma unroll
  for (int nt = 0; nt < 8; ++nt) {
    const float bv = bias[nt * 16 + m];
#pragma unroll
    for (int r = 0; r < 8; ++r) acc[nt][r] = bv;
  }

  const float* ap = As + (wave * 16 + m) * AS_STR + kh * 2;
  const float* wp = Ws + 2 * (kh * WP_STR + m);   // k2 = 2*ks + kh

  for (int ks = 0; ks < 32; ++ks) {
    v2f a;
    a.x = ap[ks * 4];
    a.y = ap[ks * 4 + 1];
    const float* wk = wp + ks * 2 * WP_STR * 2;
#pragma unroll
    for (int nt = 0; nt < 8; ++nt) {
      v2f b = *(const v2f*)(wk + nt * 32);        // one aligned 8B LDS load
      acc[nt] = __builtin_amdgcn_wmma_f32_16x16x4_f32(false, a, false, b,
                                                      (short)0, acc[nt], false, false);
    }
  }

  const int rb = rowBase + wave * 16 + kh * 8;
#pragma unroll
  for (int nt = 0; nt < 8; ++nt) {
    const int col = nt * 16 + m;
#pragma unroll
    for (int r = 0; r < 8; ++r) {
      int gr = rb + r;
      if (gr < nrows) {
        float v = acc[nt][r];
        if (EPI_RELU) v = fmaxf(v, 0.f);
        C[(size_t)gr * D + col] = v;
      }
    }
  }
}

// ---------------- host orchestration ----------------

extern "C" void kernel_launch(void* const* d_in, const int* in_sizes, int n_in,
                              void* d_out, int out_size, void* d_ws, size_t ws_size,
                              hipStream_t stream) {
  const float* h = (const float*)d_in[0];
  const int* src = (const int*)d_in[1];
  const int* dst = (const int*)d_in[2];
  const float* lin1_w = (const float*)d_in[19];
  const float* lin1_b = (const float*)d_in[20];
  const float* lin2_w = (const float*)d_in[21];
  const float* lin2_b = (const float*)d_in[22];

  const int N = in_sizes[0] / D;
  const int E = in_sizes[1];
  const long n4 = (long)N * (D / 4);

  float* Q = (float*)d_ws;                    // N*D scratch
  float* stats = Q + (size_t)N * D;           // [sum 128 | sumsq 128 | scale 128 | shift 128]
  float* scsh = stats + 256;
  float* P = (float*)d_out;                   // N*D, used as ping-pong scratch too

  const dim3 blk(256);
  const int gemmGrid = (N + 127) / 128;
  const size_t ldsBytes = (size_t)(64 * WP_STR * 2 + D * AS_STR) * sizeof(float);
  const int ewGrid = 2048;
  const int statGrid = 1024;
  const float inv_n = 1.0f / (float)N;

  for (int i = 0; i < 2; ++i) {
    const float* hin = (i == 0) ? h : P;
    const float* w0 = (const float*)d_in[3 + 8 * i];
    const float* b0 = (const float*)d_in[4 + 8 * i];
    const float* mg = (const float*)d_in[5 + 8 * i];
    const float* mb = (const float*)d_in[6 + 8 * i];
    const float* w1 = (const float*)d_in[7 + 8 * i];
    const float* b1 = (const float*)d_in[8 + 8 * i];
    const float* og = (const float*)d_in[9 + 8 * i];
    const float* ob = (const float*)d_in[10 + 8 * i];

    // Z = hin + scatter_sum(hin[src] -> dst)      (Z lives in Q)
    copy4_kernel<<<ewGrid, blk, 0, stream>>>((const float4*)hin, (float4*)Q, n4);
    edge_scatter_kernel<<<(E + 7) / 8, blk, 0, stream>>>(hin, src, dst, Q, E);

    // T = Z @ w0 + b0                             (T lives in P)
    gemm128_wmma<false, false><<<gemmGrid, blk, ldsBytes, stream>>>(Q, w0, b0, nullptr, P, N);

    // BN stats on T -> folded scale/shift
    zero_stats_kernel<<<1, blk, 0, stream>>>(stats);
    colstats_kernel<<<statGrid, blk, 0, stream>>>(P, stats, N);
    finalize_bn_kernel<<<1, 128, 0, stream>>>(stats, mg, mb, scsh, inv_n);

    // Z2 = relu(BN(T)) @ w1 + b1                  (Z2 lives in Q)
    gemm128_wmma<true, false><<<gemmGrid, blk, ldsBytes, stream>>>(P, w1, b1, scsh, Q, N);

    // outer BN stats on Z2
    zero_stats_kernel<<<1, blk, 0, stream>>>(stats);
    colstats_kernel<<<statGrid, blk, 0, stream>>>(Q, stats, N);
    finalize_bn_kernel<<<1, 128, 0, stream>>>(stats, og, ob, scsh, inv_n);

    // h_next = relu(BN(Z2))                       (back into P)
    bnrelu_kernel<<<ewGrid, blk, 0, stream>>>(Q, scsh, P, n4);
  }

  // r = relu(h2 @ lin1_w + lin1_b)                (r lives in Q)
  gemm128_wmma<false, true><<<gemmGrid, blk, ldsBytes, stream>>>(P, lin1_w, lin1_b, nullptr, Q, N);
  // out = r @ lin2_w + lin2_b                     (final -> d_out)
  gemm128_wmma<false, false><<<gemmGrid, blk, ldsBytes, stream>>>(Q, lin2_w, lin2_b, nullptr, P, N);
}